// Model_80238579024429
// MI455X (gfx1250) — compile-verified
//
#include <hip/hip_runtime.h>
#include <hip/hip_bf16.h>
#include <float.h>
#include <math.h>

#define V_    8000
#define D_    300
#define B_    128
#define L_    300
#define NGRAM_ 3
#define C_    54
#define CP_   64   // padded N for WMMA tiles

typedef __attribute__((ext_vector_type(2))) float v2f;
typedef __attribute__((ext_vector_type(8))) float v8f;

// ---------------------------------------------------------------------------
// Kernel 1: gather initial node embeddings, edge weights, eta per (b,l)
// ---------------------------------------------------------------------------
__global__ void k_gather(const float* __restrict__ node_embed,
                         const float* __restrict__ node_eta,
                         const float* __restrict__ edge_w,
                         const int*   __restrict__ docs,
                         const int*   __restrict__ edges_matrix,
                         float* __restrict__ h0,
                         float* __restrict__ wbuf,
                         float* __restrict__ etabuf) {
    const int bl = blockIdx.x;            // 0 .. B*L-1
    const int b  = bl / L_;
    const int l  = bl % L_;
    const int word = docs[bl];

    if (threadIdx.x < 7) {
        int nb = l + (int)threadIdx.x - NGRAM_;
        nb = nb < 0 ? 0 : (nb > L_ - 1 ? L_ - 1 : nb);        // clipped (matches jnp.clip)
        const int src = docs[b * L_ + nb];
        const int eid = edges_matrix[(size_t)word * V_ + src];
        wbuf[(size_t)bl * 7 + threadIdx.x] = edge_w[eid];
    }
    if (threadIdx.x == 7) {
        etabuf[bl] = node_eta[word];
    }

    const float* __restrict__ erow = node_embed + (size_t)word * D_;
    float* __restrict__ hrow = h0 + (size_t)bl * D_;
    for (int d = threadIdx.x; d < D_; d += blockDim.x) {
        hrow[d] = erow[d];
    }
}

// ---------------------------------------------------------------------------
// Kernel 2: one round of windowed weighted-max message passing
//   msg[b,l,d] = max_k valid(l,k) ? w[b,l,k] * h[b, l+k-3, d] : -FLT_MAX
//   hnew       = eta*h + (1-eta)*msg
// ---------------------------------------------------------------------------
__global__ void k_msgpass(const float* __restrict__ hprev,
                          const float* __restrict__ wbuf,
                          const float* __restrict__ etabuf,
                          float* __restrict__ hnew) {
    const int bl = blockIdx.x;            // 0 .. B*L-1
    const int l  = bl % L_;

    __shared__ float s_w[8];
    __shared__ float s_eta;
    if (threadIdx.x < 7) s_w[threadIdx.x] = wbuf[(size_t)bl * 7 + threadIdx.x];
    if (threadIdx.x == 7) s_eta = etabuf[bl];
    __syncthreads();

    const float et = s_eta;
    const float* __restrict__ hbase = hprev + (size_t)(bl - l) * D_;   // row b, l=0
    const float* __restrict__ hrow  = hprev + (size_t)bl * D_;
    float* __restrict__ orow = hnew + (size_t)bl * D_;

    for (int d = threadIdx.x; d < D_; d += blockDim.x) {
        float m = -FLT_MAX;
        #pragma unroll
        for (int k = 0; k < 7; ++k) {
            const int nb = l + k - NGRAM_;
            if (nb >= 0 && nb < L_) {                 // uniform per block -> scalar branch
                m = fmaxf(m, s_w[k] * hbase[(size_t)nb * D_ + d]);
            }
        }
        const float hp = hrow[d];
        orow[d] = et * hp + (1.0f - et) * m;
    }
}

// ---------------------------------------------------------------------------
// Kernel 3: pool over L, ReLU -> x (B, D)
// ---------------------------------------------------------------------------
__global__ void k_pool_relu(const float* __restrict__ h, float* __restrict__ x) {
    const int idx = blockIdx.x * blockDim.x + threadIdx.x;   // over B*D
    if (idx >= B_ * D_) return;
    const int b = idx / D_;
    const int d = idx % D_;
    const float* __restrict__ base = h + (size_t)b * L_ * D_ + d;
    float s = 0.0f;
    for (int l = 0; l < L_; ++l) s += base[(size_t)l * D_];
    x[idx] = fmaxf(s, 0.0f);
}

// ---------------------------------------------------------------------------
// Kernel 4: batch-norm over batch axis (block per feature d, thread per b)
// Two-pass mean/var, then normalize + affine -> xn (B, D)
// ---------------------------------------------------------------------------
__global__ void k_bn(const float* __restrict__ x,
                     const float* __restrict__ gamma,
                     const float* __restrict__ beta,
                     float* __restrict__ xn) {
    const int d = blockIdx.x;             // 0..D-1
    const int b = threadIdx.x;            // 0..127
    const float v = x[(size_t)b * D_ + d];

    __shared__ float red[B_];
    __shared__ float s_mean, s_inv;

    red[b] = v;
    __syncthreads();
    for (int s = B_ / 2; s > 0; s >>= 1) {
        if (b < s) red[b] += red[b + s];
        __syncthreads();
    }
    if (b == 0) s_mean = red[0] * (1.0f / B_);
    __syncthreads();
    const float mean = s_mean;
    const float dv = v - mean;
    red[b] = dv * dv;
    __syncthreads();
    for (int s = B_ / 2; s > 0; s >>= 1) {
        if (b < s) red[b] += red[b + s];
        __syncthreads();
    }
    if (b == 0) s_inv = 1.0f / sqrtf(red[0] * (1.0f / B_) + 1e-5f);
    __syncthreads();

    xn[(size_t)b * D_ + d] = dv * s_inv * gamma[d] + beta[d];
}

// ---------------------------------------------------------------------------
// Kernel 5: pad (54 -> 64 cols, zero-fill) and transpose W into Wt[CP_][D_],
// pad bias -> bpad[CP_]. Removes all predication from the GEMM K-loop and
// makes B-fragment K-pairs contiguous (single b64 load).
// ---------------------------------------------------------------------------
__global__ void k_prep_w(const float* __restrict__ Wm,
                         const float* __restrict__ bias,
                         float* __restrict__ Wt,
                         float* __restrict__ bpad) {
    const int idx = blockIdx.x * blockDim.x + threadIdx.x;   // over CP_*D_
    if (idx < CP_ * D_) {
        const int n = idx / D_;
        const int k = idx % D_;
        Wt[idx] = (n < C_) ? Wm[(size_t)k * C_ + n] : 0.0f;
    }
    if (idx < CP_) {
        bpad[idx] = (idx < C_) ? bias[idx] : 0.0f;
    }
}

// ---------------------------------------------------------------------------
// Kernel 6: fp32 WMMA GEMM  out = sigmoid(xn(128x300) @ W(300x54) + b)
// One block, 1024 threads = 32 waves. Wave w -> M-tile = w/4, N-tile = w%4.
// V_WMMA_F32_16X16X4_F32; 75 branch-free K-steps of K=4 (b64 loads for A, B).
//
// ISA 7.12.2 32-bit layouts:
//   A 16x4:  lanes 0-15: M=lane,    v0=K0, v1=K1 ; lanes 16-31: M=lane-16, v0=K2, v1=K3
//   B 4x16:  lanes 0-15: N=lane,    v0=K0, v1=K1 ; lanes 16-31: N=lane-16, v0=K2, v1=K3
//   C/D:     vgpr r: lanes 0-15 -> (M=r, N=lane); lanes 16-31 -> (M=r+8, N=lane-16)
// ---------------------------------------------------------------------------
__global__ void k_gemm_sigmoid(const float* __restrict__ xn,
                               const float* __restrict__ Wt,    // (CP_, D_) transposed+padded
                               const float* __restrict__ bpad,  // (CP_)
                               float* __restrict__ out) {
    const int wave = threadIdx.x >> 5;
    const int lane = threadIdx.x & 31;
    const int mt = wave >> 2;             // 0..7
    const int nt = wave & 3;              // 0..3
    const int m0 = mt * 16;
    const int n0 = nt * 16;
    const int lm = lane & 15;
    const bool hi = lane >= 16;           // lanes 16-31 carry K+2, K+3

    const int arow = m0 + lm;             // A row for this lane
    const int bcol = n0 + lm;             // B col for this lane

    const float* __restrict__ arow_p = xn + (size_t)arow * D_;
    const float* __restrict__ bcol_p = Wt + (size_t)bcol * D_;
    const int koff = hi ? 2 : 0;

    v8f acc = {};
    for (int k = 0; k < D_; k += 4) {
        const int ka = k + koff;          // even -> 8B-aligned v2f loads
        const v2f a  = *(const v2f*)(arow_p + ka);
        const v2f bb = *(const v2f*)(bcol_p + ka);
        acc = __builtin_amdgcn_wmma_f32_16x16x4_f32(
            /*neg_a=*/false, a, /*neg_b=*/false, bb,
            /*c_mod=*/(short)0, acc, /*reuse_a=*/false, /*reuse_b=*/false);
    }

    const float bv = bpad[bcol];
    #pragma unroll
    for (int r = 0; r < 8; ++r) {
        const int orow = m0 + r + (hi ? 8 : 0);
        const int ocol = n0 + lm;
        if (ocol < C_) {                  // store-side predication only (post-WMMA)
            const float z = acc[r] + bv;
            out[(size_t)orow * C_ + ocol] = 1.0f / (1.0f + expf(-z));
        }
    }
}

// ---------------------------------------------------------------------------
// Launch
// ---------------------------------------------------------------------------
extern "C" void kernel_launch(void* const* d_in, const int* in_sizes, int n_in,
                              void* d_out, int out_size, void* d_ws, size_t ws_size,
                              hipStream_t stream) {
    const float* node_embed   = (const float*)d_in[0];   // V*D
    const float* node_eta     = (const float*)d_in[1];   // V
    const float* edge_w       = (const float*)d_in[2];   // E_NUM
    const float* bn_gamma     = (const float*)d_in[3];   // D
    const float* bn_beta      = (const float*)d_in[4];   // D
    const float* Wm           = (const float*)d_in[5];   // D*C
    const float* bias         = (const float*)d_in[6];   // C
    const int*   docs         = (const int*)d_in[7];     // B*L
    const int*   edges_matrix = (const int*)d_in[8];     // V*V
    float* out = (float*)d_out;

    // Workspace layout (floats; all offsets even -> 8B alignment preserved)
    const size_t n_h   = (size_t)B_ * L_ * D_;           // 11,520,000
    const size_t n_w   = (size_t)B_ * L_ * 7;            //    268,800
    const size_t n_bl  = (size_t)B_ * L_;                //     38,400
    const size_t n_bd  = (size_t)B_ * D_;                //     38,400
    const size_t n_wt  = (size_t)CP_ * D_;               //     19,200
    float* ws   = (float*)d_ws;
    float* h0   = ws;
    float* h1   = h0 + n_h;
    float* wbuf = h1 + n_h;
    float* etab = wbuf + n_w;
    float* xbuf = etab + n_bl;
    float* xn   = xbuf + n_bd;
    float* Wt   = xn + n_bd;
    float* bpad = Wt + n_wt;

    const int BL = B_ * L_;

    k_gather<<<BL, 128, 0, stream>>>(node_embed, node_eta, edge_w, docs,
                                     edges_matrix, h0, wbuf, etab);

    // W prep is independent of the message-passing chain; issue early
    k_prep_w<<<(CP_ * D_ + 255) / 256, 256, 0, stream>>>(Wm, bias, Wt, bpad);

    // T = 2 rounds of message passing (ping-pong h0 -> h1 -> h0)
    k_msgpass<<<BL, 128, 0, stream>>>(h0, wbuf, etab, h1);
    k_msgpass<<<BL, 128, 0, stream>>>(h1, wbuf, etab, h0);

    k_pool_relu<<<(B_ * D_ + 255) / 256, 256, 0, stream>>>(h0, xbuf);

    k_bn<<<D_, B_, 0, stream>>>(xbuf, bn_gamma, bn_beta, xn);

    k_gemm_sigmoid<<<1, 1024, 0, stream>>>(xn, Wt, bpad, out);
}